// PropositionInputModule_59665685676093
// MI455X (gfx1250) — compile-verified
//
#include <hip/hip_runtime.h>
#include <stdint.h>

// Problem constants from the reference.
#define HD 256   // H
#define AD 4096  // A
#define BD 32    // B
#define KD 16    // K

typedef __attribute__((ext_vector_type(4))) unsigned int u32x4;
typedef __attribute__((ext_vector_type(8))) int          i32x8;
typedef __attribute__((ext_vector_type(4))) int          i32x4;

__global__ __launch_bounds__(256) void gather_max_tdm(const float* __restrict__ x,
                                                      const int* __restrict__ idx,
                                                      float* __restrict__ out) {
  // Per-wave double-buffered gather tiles: 8 waves x 2 x (16 rows x 256 f32)
  // = 256 KB, plus 8 KB fold scratch. 264 KB of the WGP's 320 KB LDS.
  __shared__ float tile[8][2][KD * HD];
  __shared__ float partial[8][HD];

  const int g  = blockIdx.x;
  const int t  = threadIdx.x;
  const int w  = t >> 5;   // wave id 0..7 (wave32)
  const int ln = t & 31;   // lane id

  // Mark LDS as written so the compiler cannot UB-fold reads of
  // TDM-written (compiler-invisible) memory.
  if (t == 0) { tile[0][0][0] = 0.0f; partial[0][0] = 0.0f; }

  // ---------------- Build the gather D# (block-uniform) ----------------
  // Groups 2+3: 16 row indices, 16-bit packed (idx values < 4096 fit).
  uint32_t iw[8];
#pragma unroll
  for (int i = 0; i < 8; ++i) {
    uint32_t lo = (uint32_t)idx[g * KD + 2 * i]     & 0xFFFFu;
    uint32_t hi = (uint32_t)idx[g * KD + 2 * i + 1] & 0xFFFFu;
    iw[i] = (uint32_t)__builtin_amdgcn_readfirstlane((int)(lo | (hi << 16)));
  }
  i32x4 g2, g3;
  g2[0] = (int)iw[0]; g2[1] = (int)iw[1]; g2[2] = (int)iw[2]; g2[3] = (int)iw[3];
  g3[0] = (int)iw[4]; g3[1] = (int)iw[5]; g3[2] = (int)iw[6]; g3[3] = (int)iw[7];

  // Group 1: data_size=4B (code 2); tensor_dim0=256; tensor_dim1=4096;
  // tile_dim0=256; tile_dim1=16 valid indices; tensor_dim0_stride=256.
  i32x8 g1;
  g1[0] = 0x00020000;        // [17:16] data_size = 2 (4 bytes)
  g1[1] = (int)(HD << 16);   // bits[63:48]  = tensor_dim0[15:0]
  g1[2] = (int)(AD << 16);   // bits[95:80]  = tensor_dim1[15:0]
  g1[3] = (int)(HD << 16);   // bits[127:112]= tile_dim0
  g1[4] = KD;                // bits[143:128]= tile_dim1 = #valid indices
  g1[5] = HD;                // tensor_dim0_stride[31:0]
  g1[6] = 0;
  g1[7] = 0;

  // Unused VADDR4 group for the 6-arg builtin form: zeros.
  i32x8 g4;
  g4[0] = 0; g4[1] = 0; g4[2] = 0; g4[3] = 0;
  g4[4] = 0; g4[5] = 0; g4[6] = 0; g4[7] = 0;

  const uint32_t ldsA = (uint32_t)(uintptr_t)(&tile[w][0][0]);
  const uint32_t ldsB = (uint32_t)(uintptr_t)(&tile[w][1][0]);

  auto issue = [&](int b, uint32_t ldsaddr) {
    const uint64_t ga = (uint64_t)(uintptr_t)(x + (uint64_t)b * (uint64_t)(AD * HD));
    u32x4 g0;
    g0[0] = 0x80000001u;                     // count=1, gather_mode=1, 16-bit idx
    g0[1] = ldsaddr;                         // LDS byte address
    g0[2] = (uint32_t)(ga & 0xFFFFFFFFu);    // global_addr[31:0]
    g0[3] = (uint32_t)((ga >> 32) & 0x01FFFFFFu) | 0x80000000u; // addr[56:32] | type=2
    __builtin_amdgcn_tensor_load_to_lds(g0, g1, g2, g3, g4, 0);
  };

  // Each wave owns batches {w, w+8, w+16, w+24}: per-wave TDM pipeline,
  // self-synchronized via its own TENSORcnt. No workgroup barriers in the loop.
  issue(w,     ldsA);
  issue(w + 8, ldsB);

  // Lane mapping: 8 h-columns per lane (two b128 LDS loads per k-row).
  const int hq = ln << 3;  // h base: 0,8,...,248

  float4 acc0, acc1;
  acc0.x = acc0.y = acc0.z = acc0.w = -3.4028234663852886e+38f;
  acc1 = acc0;

#pragma unroll
  for (int i = 0; i < 4; ++i) {
    if (i < 3) __builtin_amdgcn_s_wait_tensorcnt(1);  // oldest DMA done (in-order)
    else       __builtin_amdgcn_s_wait_tensorcnt(0);

    const float* buf = &tile[w][i & 1][0];
#pragma unroll
    for (int k = 0; k < KD; ++k) {
      const float4 v0 = *(const float4*)&buf[k * HD + hq];      // ds_load_b128
      const float4 v1 = *(const float4*)&buf[k * HD + hq + 4];  // ds_load_b128
      acc0.x = fmaxf(acc0.x, v0.x); acc0.y = fmaxf(acc0.y, v0.y);
      acc0.z = fmaxf(acc0.z, v0.z); acc0.w = fmaxf(acc0.w, v0.w);
      acc1.x = fmaxf(acc1.x, v1.x); acc1.y = fmaxf(acc1.y, v1.y);
      acc1.z = fmaxf(acc1.z, v1.z); acc1.w = fmaxf(acc1.w, v1.w);
    }
    // Refill the buffer just consumed (DScnt waits for the v_max consumers
    // precede this in program order, so no WAR on the LDS buffer).
    if (i + 2 < 4) issue(w + 8 * (i + 2), (i & 1) ? ldsB : ldsA);
  }

  // Cross-wave fold: 8 wave-partials -> one 256-wide result.
  *(float4*)&partial[w][hq]     = acc0;
  *(float4*)&partial[w][hq + 4] = acc1;
  __syncthreads();

  float m = partial[0][t];
#pragma unroll
  for (int j = 1; j < 8; ++j) m = fmaxf(m, partial[j][t]);
  out[(size_t)g * HD + t] = m;
}

extern "C" void kernel_launch(void* const* d_in, const int* in_sizes, int n_in,
                              void* d_out, int out_size, void* d_ws, size_t ws_size,
                              hipStream_t stream) {
  const float* x   = (const float*)d_in[0];
  const int*   idx = (const int*)d_in[1];
  float*       out = (float*)d_out;

  const int G = in_sizes[1] / KD;  // 512
  dim3 grid(G);
  dim3 block(256);
  gather_max_tdm<<<grid, block, 0, stream>>>(x, idx, out);
}